// WaveletDenoising_5291399708730
// MI455X (gfx1250) — compile-verified
//
#include <hip/hip_runtime.h>
#include <cstdint>

// ---------------- CDNA5 feature detection (device pass only) ----------------
#if defined(__gfx1250__) && __has_builtin(__builtin_amdgcn_wmma_f32_16x16x4_f32)
#define USE_WMMA 1
#else
#define USE_WMMA 0
#endif

#if defined(__gfx1250__) && __has_builtin(__builtin_amdgcn_tensor_load_to_lds) && \
    __has_builtin(__builtin_amdgcn_tensor_store_from_lds) &&                       \
    __has_builtin(__builtin_amdgcn_s_wait_tensorcnt)
#define USE_TDM 1
#else
#define USE_TDM 0
#endif

typedef float v2f  __attribute__((ext_vector_type(2)));
typedef float v4f  __attribute__((ext_vector_type(4)));
typedef float v8f  __attribute__((ext_vector_type(8)));
typedef unsigned int u32x4 __attribute__((ext_vector_type(4)));
typedef int   i32x4 __attribute__((ext_vector_type(4)));
typedef int   i32x8 __attribute__((ext_vector_type(8)));

#define TPB 256
#define S0  8192          // signal length
#define P1  4099          // level-1 coeff length = (S0+5)/2 + 1
#define P2  2053          // level-2 coeff length = (P1+5)/2 + 1
#define LY1 4100          // first reconstruction length = 2*P2 - 6
#define KTH 2049          // 0-based median index of P1 elements
#define MEDPAD (17 * TPB) // d1 padded with +INF to 4352 -> 17 regs/thread

// dynamic-LDS layout (floats); every region 16B aligned, tile-rounded sizes
#define A1_ALLOC 4112     // ceil(P1/16)*16
#define D1_ALLOC 4352     // MEDPAD
#define A2_ALLOC 2064     // ceil(P2/16)*16
#define D2_ALLOC 2064
#define OFF_SIG 0
#define OFF_A1  8192
#define OFF_D1  (OFF_A1 + A1_ALLOC)
#define OFF_A2  (OFF_D1 + D1_ALLOC)
#define OFF_D2  (OFF_A2 + A2_ALLOC)
#define SMEM_FLOATS (OFF_D2 + D2_ALLOC)   // 20784 floats = 83136 B

// db4 reconstruction filters; analysis lo/hi are correlation with these
// (DEC_LO[7-t] == REC_LO[t], DEC_HI[7-t] == REC_HI[t] for this orthogonal bank)
static constexpr float c_rlo[8] = {
  0.23037781330885523f,  0.7148465705525415f,  0.6308807679295904f,
 -0.02798376941698385f, -0.18703481171888114f, 0.030841381835986965f,
  0.032883011666982945f, -0.010597401784997278f };
static constexpr float c_rhi[8] = {
 -0.010597401784997278f, -0.032883011666982945f, 0.030841381835986965f,
  0.18703481171888114f,  -0.02798376941698385f, -0.6308807679295904f,
  0.7148465705525415f,   -0.23037781330885523f };

// symmetric-extension read: logical index i in [-6, 2S), reflect at both ends
__device__ __forceinline__ float extv(const float* __restrict s, int i, int S) {
  i = (i < 0)  ? (-1 - i)        : i;
  i = (i >= S) ? (2 * S - 1 - i) : i;
  return s[i];
}

// --------------------------- analysis (DWT) step ----------------------------
// a[p] = sum_t ext(2p+t-6)*REC_LO[t] ; d[p] = sum_t ext(2p+t-6)*REC_HI[t]
// Output arrays must be allocated to ceil(P/16)*16 (tail stores unconditional).
__device__ void dwt_level(const float* __restrict src, int S,
                          float* __restrict a, float* __restrict d, int P)
{
  const int tid = threadIdx.x;
#if USE_WMMA
  const int lane = tid & 31;
  const int wave = tid >> 5;
  const int m    = lane & 15;   // A row (output position) / B-D column
  const int kh   = lane >> 4;   // K-half per documented 16x4 f32 A striping
  // B is tile-invariant: column 0 = REC_LO taps, column 1 = REC_HI taps.
  // Assumed B striping mirrors A: VGPR0 -> K=2*kh, VGPR1 -> K=2*kh+1, N=lane&15.
  // constexpr indices fold to inline-constant cndmask selects (no memory).
  v2f b_lo, b_hi;
  b_lo.x = (m == 0) ? (kh ? c_rlo[2] : c_rlo[0]) : (m == 1) ? (kh ? c_rhi[2] : c_rhi[0]) : 0.0f;
  b_lo.y = (m == 0) ? (kh ? c_rlo[3] : c_rlo[1]) : (m == 1) ? (kh ? c_rhi[3] : c_rhi[1]) : 0.0f;
  b_hi.x = (m == 0) ? (kh ? c_rlo[6] : c_rlo[4]) : (m == 1) ? (kh ? c_rhi[6] : c_rhi[4]) : 0.0f;
  b_hi.y = (m == 0) ? (kh ? c_rlo[7] : c_rlo[5]) : (m == 1) ? (kh ? c_rhi[7] : c_rhi[5]) : 0.0f;

  for (int p0 = wave * 16; p0 < P; p0 += 16 * (TPB / 32)) {  // uniform per wave
    const int i0 = 2 * (p0 + m) - 6 + 2 * kh;                // even -> 8B aligned
    v2f a_lo, a_hi;
    if (p0 >= 3 && 2 * p0 + 31 < S) {      // wave-uniform interior fast path
      a_lo = *(const v2f*)(src + i0);
      a_hi = *(const v2f*)(src + i0 + 4);
    } else {                                // boundary tiles: reflected reads
      a_lo.x = extv(src, i0,     S);
      a_lo.y = extv(src, i0 + 1, S);
      a_hi.x = extv(src, i0 + 4, S);
      a_hi.y = extv(src, i0 + 5, S);
    }
    v8f acc = {0.f, 0.f, 0.f, 0.f, 0.f, 0.f, 0.f, 0.f};
    acc = __builtin_amdgcn_wmma_f32_16x16x4_f32(false, a_lo, false, b_lo,
                                                (short)0, acc, false, false);
    acc = __builtin_amdgcn_wmma_f32_16x16x4_f32(false, a_hi, false, b_hi,
                                                (short)0, acc, false, false);
    // D layout: VGPR r, lanes 0-15 -> M=r ; lanes 16-31 -> M=r+8 ; N=lane&15
    // Each holding lane owns 8 contiguous outputs -> two aligned b128 stores.
    if (m < 2) {
      float* __restrict dst = ((m == 0) ? a : d) + p0 + 8 * kh;
      v4f lo4 = {acc[0], acc[1], acc[2], acc[3]};
      v4f hi4 = {acc[4], acc[5], acc[6], acc[7]};
      *(v4f*)(dst)     = lo4;
      *(v4f*)(dst + 4) = hi4;
    }
  }
#else
  for (int p = tid; p < P; p += TPB) {
    float sa = 0.0f, sd = 0.0f;
#pragma unroll
    for (int t = 0; t < 8; ++t) {
      const float v = extv(src, 2 * p - 6 + t, S);
      sa = fmaf(v, c_rlo[t], sa);
      sd = fmaf(v, c_rhi[t], sd);
    }
    a[p] = sa; d[p] = sd;
  }
#endif
}

// ------------------------- synthesis (iDWT) step ----------------------------
// Two consecutive outputs share the same 4 a-taps + 4 d-taps (base = p>>1);
// index ranges verified in-bounds. ly is even; stores are aligned b64.
__device__ void idwt_level(const float* __restrict a, const float* __restrict d,
                           float* __restrict y, int ly)
{
  const int half = ly >> 1;
  for (int i = threadIdx.x; i < half; i += TPB) {
    const float a0 = a[i], a1v = a[i + 1], a2v = a[i + 2], a3v = a[i + 3];
    const float d0 = d[i], d1v = d[i + 1], d2v = d[i + 2], d3v = d[i + 3];
    float ev = fmaf(a0, c_rlo[6], fmaf(d0, c_rhi[6],
               fmaf(a1v, c_rlo[4], fmaf(d1v, c_rhi[4],
               fmaf(a2v, c_rlo[2], fmaf(d2v, c_rhi[2],
               fmaf(a3v, c_rlo[0], d3v * c_rhi[0])))))));
    float od = fmaf(a0, c_rlo[7], fmaf(d0, c_rhi[7],
               fmaf(a1v, c_rlo[5], fmaf(d1v, c_rhi[5],
               fmaf(a2v, c_rlo[3], fmaf(d2v, c_rhi[3],
               fmaf(a3v, c_rlo[1], d3v * c_rhi[1])))))));
    v2f o = {ev, od};
    *(v2f*)(y + 2 * i) = o;
  }
}

__device__ void soft_arr(float* __restrict v, int n, float t) {
  for (int i = threadIdx.x; i < n; i += TPB) {
    const float x = v[i];
    v[i] = copysignf(fmaxf(fabsf(x) - t, 0.0f), x);
  }
}

// exact k-th order statistic of |v| via bisection on the float bit pattern.
// d1 must be +INF padded to MEDPAD; the 17 values/thread live in registers,
// so each of the ~31 rounds is pure VALU + one LDS atomic per wave.
__device__ float median_abs_reg(const float* __restrict d1, int* s_cnt)
{
  const int tid = threadIdx.x;
  unsigned av[17];
#pragma unroll
  for (int j = 0; j < 17; ++j)
    av[j] = __float_as_uint(d1[tid + j * TPB]) & 0x7fffffffu;

  unsigned lo = 0u, hi = 0x7f800000u;
  while (lo < hi) {                      // block-uniform loop (~31 iterations)
    const unsigned mid = (lo + hi) >> 1;
    if (tid == 0) *s_cnt = 0;
    __syncthreads();
    int c = 0;
#pragma unroll
    for (int j = 0; j < 17; ++j) c += (av[j] <= mid) ? 1 : 0;
#pragma unroll
    for (int off = 16; off > 0; off >>= 1) c += __shfl_down(c, off, 32);
    if ((tid & 31) == 0) atomicAdd(s_cnt, c);
    __syncthreads();
    const int cnt = *s_cnt;
    __syncthreads();
    if (cnt > KTH) hi = mid; else lo = mid + 1;
  }
  return __uint_as_float(lo);
}

// ---------------- Tensor Data Mover: strided signal gather/scatter ----------
#if USE_TDM
// 2D D#: rows y=0..8191 with stride 64 elements, 1 element/row, 4-byte data.
__device__ __forceinline__ void tdm_desc(u32x4& g0, i32x8& g1,
                                         unsigned lds_byte, unsigned long long ga)
{
  g0[0] = 1u;                                        // count=1, user D#, no gather
  g0[1] = lds_byte;                                  // lds_addr
  g0[2] = (unsigned)(ga & 0xffffffffull);            // global_addr[31:0]
  g0[3] = (unsigned)((ga >> 32) & 0x01ffffffull)     // global_addr[56:32]
        | 0x80000000u;                               // type=2 ("image")
  g1[0] = 0x00020000;            // workgroup_mask=0, data_size=2 (4 bytes)
  g1[1] = (int)(64u << 16);      // tensor_dim0 = 64      (bits 79:48 low half)
  g1[2] = (int)(8192u << 16);    // tensor_dim1 = 8192    (bits 111:80 low half)
  g1[3] = (int)(1u << 16);       // tile_dim0 = 1         (bits 127:112)
  g1[4] = 8192;                  // tile_dim1 = 8192      (bits 143:128)
  g1[5] = 64;                    // tensor_dim0_stride = 64 elements
  g1[6] = 0;                     // stride0 hi / stride1 lo
  g1[7] = 0;                     // stride1 hi
}
__device__ __forceinline__ void tdm_load_signal(const float* g, const float* lds)
{
  u32x4 g0; i32x8 g1;
  tdm_desc(g0, g1, (unsigned)(uintptr_t)lds, (unsigned long long)(uintptr_t)g);
  i32x4 z4 = {0, 0, 0, 0};
#if __clang_major__ >= 23
  i32x8 z8 = {0, 0, 0, 0, 0, 0, 0, 0};
  __builtin_amdgcn_tensor_load_to_lds(g0, g1, z4, z4, z8, 0);
#else
  __builtin_amdgcn_tensor_load_to_lds(g0, g1, z4, z4, 0);
#endif
}
__device__ __forceinline__ void tdm_store_signal(float* g, const float* lds)
{
  u32x4 g0; i32x8 g1;
  tdm_desc(g0, g1, (unsigned)(uintptr_t)lds, (unsigned long long)(uintptr_t)g);
  i32x4 z4 = {0, 0, 0, 0};
#if __clang_major__ >= 23
  i32x8 z8 = {0, 0, 0, 0, 0, 0, 0, 0};
  __builtin_amdgcn_tensor_store_from_lds(g0, g1, z4, z4, z8, 0);
#else
  __builtin_amdgcn_tensor_store_from_lds(g0, g1, z4, z4, 0);
#endif
}
#endif // USE_TDM

// --------------------------------- kernel -----------------------------------
__global__ __launch_bounds__(TPB)
void wavelet_denoise_kernel(const float* __restrict in, float* __restrict out)
{
  extern __shared__ float sm[];
  float* sig = sm + OFF_SIG;   // signal, later final reconstruction
  float* a1  = sm + OFF_A1;    // level-1 approx, later rec1
  float* d1  = sm + OFF_D1;    // level-1 detail (+INF padded for median)
  float* a2  = sm + OFF_A2;    // level-2 approx
  float* d2  = sm + OFF_D2;    // level-2 detail
  __shared__ int s_cnt;

  const int n = blockIdx.x;          // signal id in [0, 2048)
  const int b = n >> 6;
  const int f = n & 63;
  const float* gsrc = in  + (size_t)b * S0 * 64 + f;
  float*       gdst = out + (size_t)b * S0 * 64 + f;

  // ---- stage signal (stride-256B gather) into LDS ----
#if USE_TDM
  if (threadIdx.x < 32) {            // wave 0 issues one TDM DMA
    tdm_load_signal(gsrc, sig);
    __builtin_amdgcn_s_wait_tensorcnt(0);
  }
#else
  for (int s = threadIdx.x; s < S0; s += TPB) sig[s] = gsrc[(size_t)s * 64];
#endif
  __syncthreads();

  dwt_level(sig, S0, a1, d1, P1);
  __syncthreads();                                  // d1 tail garbage now stable
  dwt_level(a1, P1, a2, d2, P2);
  // overwrite d1 tail (incl. tile-store garbage at [P1, 4112)) with +INF pads
  for (int i = P1 + threadIdx.x; i < MEDPAD; i += TPB)
    d1[i] = __uint_as_float(0x7f800000u);
  __syncthreads();

  const float med = median_abs_reg(d1, &s_cnt);
  const float thr = (med / 0.6745f) * 4.2452126f;   // sqrt(2*ln(8192))

  soft_arr(a2, P2, thr);
  soft_arr(d2, P2, thr);
  soft_arr(d1, P1, thr);
  __syncthreads();

  idwt_level(a2, d2, a1, LY1);       // rec1 (4100) overwrites dead a1
  __syncthreads();
  idwt_level(a1, d1, sig, S0);       // final reconstruction (8192)
  __syncthreads();

  // ---- scatter result back (stride-256B) ----
#if USE_TDM
  if (threadIdx.x < 32) {
    tdm_store_signal(gdst, sig);
    __builtin_amdgcn_s_wait_tensorcnt(0);
  }
#else
  for (int s = threadIdx.x; s < S0; s += TPB) gdst[(size_t)s * 64] = sig[s];
#endif
}

// -------------------------------- launcher ----------------------------------
extern "C" void kernel_launch(void* const* d_in, const int* in_sizes, int n_in,
                              void* d_out, int out_size, void* d_ws, size_t ws_size,
                              hipStream_t stream)
{
  (void)in_sizes; (void)n_in; (void)d_ws; (void)ws_size; (void)out_size;
  const float* x = (const float*)d_in[0];
  float*       y = (float*)d_out;

  const size_t smem = (size_t)SMEM_FLOATS * sizeof(float);  // ~83 KB (<320 KB WGP LDS)
  // opt in to >64KB dynamic LDS where the runtime requires it (no-op otherwise)
  (void)hipFuncSetAttribute((const void*)wavelet_denoise_kernel,
                            hipFuncAttributeMaxDynamicSharedMemorySize, (int)smem);

  hipLaunchKernelGGL(wavelet_denoise_kernel, dim3(2048), dim3(TPB), smem, stream,
                     x, y);
}